// W8A16Linear_35811437314659
// MI455X (gfx1250) — compile-verified
//
#include <hip/hip_runtime.h>
#include <hip/hip_bf16.h>
#include <stdint.h>

#define IN_F   8192
#define OUT_F  8192
#define BATCH  16

typedef __attribute__((ext_vector_type(2))) float v2f;
typedef __attribute__((ext_vector_type(4))) float v4f;
typedef __attribute__((ext_vector_type(8))) float v8f;

__device__ __forceinline__ float quantize1(float w, float inv_s) {
    // zp == 0 (proved from the reference's symmetric min/max), RTE rounding
    float q = rintf(w * inv_s);
    return fminf(fmaxf(q, -128.0f), 127.0f);
}

__device__ __forceinline__ void unpack_i8pair(uint16_t u, v2f& a) {
    a[0] = (float)(int)(int8_t)(u & 0xFF);
    a[1] = (float)(int)(int8_t)(u >> 8);
}

// ---------------------------------------------------------------------------
// Kernel 1: one workgroup per weight row.
//  - stream the row with non-temporal loads (don't thrash L2 with 268 MB)
//  - LDS max-reduce -> scale
//  - optionally write the int8 quantized row (64 MB total; L2-resident)
// ---------------------------------------------------------------------------
__global__ void __launch_bounds__(256) quant_rows_kernel(
    const float* __restrict__ W, float* __restrict__ scales,
    int8_t* __restrict__ wq, int store_q)
{
    const int row = blockIdx.x;
    const int tid = threadIdx.x;
    const float* rp = W + (size_t)row * IN_F;

    v4f v[8];
    float mx = 0.0f;
#pragma unroll
    for (int c = 0; c < 8; ++c) {
        const v4f* p = (const v4f*)(rp + c * 1024 + tid * 4);
        v[c] = __builtin_nontemporal_load(p);   // TH=NT: keep L2 for the int8 output
        mx = fmaxf(mx, fmaxf(fmaxf(fabsf(v[c].x), fabsf(v[c].y)),
                             fmaxf(fabsf(v[c].z), fabsf(v[c].w))));
    }

    __shared__ float sm[256];
    sm[tid] = mx;
    __syncthreads();
    for (int s = 128; s > 0; s >>= 1) {
        if (tid < s) sm[tid] = fmaxf(sm[tid], sm[tid + s]);
        __syncthreads();
    }
    mx = sm[0];

    // denom = 2*max ; scale = denom/255 (or 1.0 if row is all zeros)
    const float scale = (mx > 0.0f) ? mx * (2.0f / 255.0f) : 1.0f;
    if (tid == 0) scales[row] = scale;

    if (store_q) {
        const float inv_s = 1.0f / scale;
        int8_t* qp = wq + (size_t)row * IN_F;
#pragma unroll
        for (int c = 0; c < 8; ++c) {
            int q0 = (int)quantize1(v[c].x, inv_s);
            int q1 = (int)quantize1(v[c].y, inv_s);
            int q2 = (int)quantize1(v[c].z, inv_s);
            int q3 = (int)quantize1(v[c].w, inv_s);
            int32_t pk = (q0 & 0xFF) | ((q1 & 0xFF) << 8) |
                         ((q2 & 0xFF) << 16) | ((q3 & 0xFF) << 24);
            *(int32_t*)(qp + c * 1024 + tid * 4) = pk;   // coalesced b32 stores
        }
    }
}

// ---------------------------------------------------------------------------
// Kernel 2 (main path): WMMA f32 16x16x4 GEMM on the int8 matrix.
// One WG per PAIR of 16-row M tiles: the x (B) fragments are loaded once and
// feed both tiles -> halves B-side L2 traffic (268 MB -> 134 MB) and gives
// 4 independent WMMA accumulator chains (2 tiles x 2 k-phases).
// 8 waves split K=8192 into 1024-wide chunks; partial C tiles reduced in LDS;
// postscale by s_row, add bias.
//
// A fragment (16x4 f32): lane (m = lane&15, hi = lane>>4) holds
//   W[m][k + hi*2 + 0], W[m][k + hi*2 + 1]    -> one u16 load (2x int8)
// B fragment (4x16 f32): lane (b = lane&15, hi) holds
//   x[b][k + hi*2 + 0],  x[b][k + hi*2 + 1]   -> one b64 load
// ---------------------------------------------------------------------------
__global__ void __launch_bounds__(256) gemm_wmma_q_kernel(
    const int8_t* __restrict__ wq, const float* __restrict__ scales,
    const float* __restrict__ x, const float* __restrict__ bias,
    float* __restrict__ out)
{
    const int tile0 = blockIdx.x * 2;
    const int tid   = threadIdx.x;
    const int wave  = tid >> 5;
    const int lane  = tid & 31;
    const int m     = lane & 15;     // A: row within tile ; B: batch column
    const int hi    = lane >> 4;
    const int kbase = wave * 1024 + hi * 2;

    const int8_t* ap0 = wq + (size_t)(tile0 * 16 + m) * IN_F + kbase;
    const int8_t* ap1 = ap0 + (size_t)16 * IN_F;            // next M tile
    const float*  bp  = x + (size_t)m * IN_F + kbase;

    v8f c00 = {0.f,0.f,0.f,0.f,0.f,0.f,0.f,0.f};  // tile0, k-phase 0
    v8f c01 = {0.f,0.f,0.f,0.f,0.f,0.f,0.f,0.f};  // tile0, k-phase 1
    v8f c10 = {0.f,0.f,0.f,0.f,0.f,0.f,0.f,0.f};  // tile1, k-phase 0
    v8f c11 = {0.f,0.f,0.f,0.f,0.f,0.f,0.f,0.f};  // tile1, k-phase 1

#pragma unroll 2
    for (int it = 0; it < 128; ++it) {            // 2 x K=4 per iteration
        v2f b0 = *(const v2f*)(bp);
        v2f b1 = *(const v2f*)(bp + 4);
        v2f a00, a01, a10, a11;
        unpack_i8pair(*(const uint16_t*)(ap0),     a00);
        unpack_i8pair(*(const uint16_t*)(ap0 + 4), a01);
        unpack_i8pair(*(const uint16_t*)(ap1),     a10);
        unpack_i8pair(*(const uint16_t*)(ap1 + 4), a11);
        c00 = __builtin_amdgcn_wmma_f32_16x16x4_f32(false, a00, false, b0,
                                                    (short)0, c00, false, false);
        c01 = __builtin_amdgcn_wmma_f32_16x16x4_f32(false, a01, false, b1,
                                                    (short)0, c01, false, false);
        c10 = __builtin_amdgcn_wmma_f32_16x16x4_f32(false, a10, false, b0,
                                                    (short)0, c10, false, false);
        c11 = __builtin_amdgcn_wmma_f32_16x16x4_f32(false, a11, false, b1,
                                                    (short)0, c11, false, false);
        ap0 += 8;
        ap1 += 8;
        bp  += 8;
    }
    c00 += c01;   // elementwise add of identically-laid-out C fragments
    c10 += c11;

    __shared__ float red[8 * 256];

    // ---- reduce + store, tile 0 then tile 1 (reuse the 8 KB LDS buffer) ----
#pragma unroll
    for (int t = 0; t < 2; ++t) {
        const v8f& acc = (t == 0) ? c00 : c10;
        const int tile = tile0 + t;
#pragma unroll
        for (int v = 0; v < 8; ++v) red[wave * 256 + v * 32 + lane] = acc[v];
        __syncthreads();
        {
            // 256 threads: each reduces one (vgpr-slot, lane) cell over 8 waves
            const int v = tid >> 5;
            const int l = tid & 31;
            float s = 0.0f;
#pragma unroll
            for (int w = 0; w < 8; ++w) s += red[w * 256 + v * 32 + l];
            // D layout: lanes 0-15: M=v ; lanes 16-31: M=v+8 ; N=l&15
            const int rl   = v + (l >> 4) * 8;
            const int b    = l & 15;
            const int orow = tile * 16 + rl;
            out[orow * BATCH + b] = scales[orow] * s + bias[orow];
        }
        __syncthreads();   // all reads done before buffer reuse
    }
}

// ---------------------------------------------------------------------------
// Kernel 2 (fallback, if ws can't hold the 64 MB int8 intermediate):
// identical WMMA structure (single tile), but A comes from fp32 W quantize-
// dequantized inline (needs only the 32 KB scales array in ws).
// ---------------------------------------------------------------------------
__global__ void __launch_bounds__(256) gemm_wmma_f_kernel(
    const float* __restrict__ W, const float* __restrict__ scales,
    const float* __restrict__ x, const float* __restrict__ bias,
    float* __restrict__ out)
{
    const int tile = blockIdx.x;
    const int tid  = threadIdx.x;
    const int wave = tid >> 5;
    const int lane = tid & 31;
    const int m    = lane & 15;
    const int hi   = lane >> 4;
    const int kbase = wave * 1024 + hi * 2;
    const int row  = tile * 16 + m;

    const float* ap = W + (size_t)row * IN_F + kbase;
    const float* bp = x + (size_t)m   * IN_F + kbase;
    const float inv_s = 1.0f / scales[row];

    v8f c0 = {0.f,0.f,0.f,0.f,0.f,0.f,0.f,0.f};
    v8f c1 = {0.f,0.f,0.f,0.f,0.f,0.f,0.f,0.f};

#pragma unroll 4
    for (int it = 0; it < 128; ++it) {
        v2f w0 = *(const v2f*)(ap);
        v2f w1 = *(const v2f*)(ap + 4);
        v2f a0, a1;
        a0[0] = quantize1(w0[0], inv_s); a0[1] = quantize1(w0[1], inv_s);
        a1[0] = quantize1(w1[0], inv_s); a1[1] = quantize1(w1[1], inv_s);
        v2f b0 = *(const v2f*)(bp);
        v2f b1 = *(const v2f*)(bp + 4);
        c0 = __builtin_amdgcn_wmma_f32_16x16x4_f32(false, a0, false, b0,
                                                   (short)0, c0, false, false);
        c1 = __builtin_amdgcn_wmma_f32_16x16x4_f32(false, a1, false, b1,
                                                   (short)0, c1, false, false);
        ap += 8;
        bp += 8;
    }
    c0 += c1;

    __shared__ float red[8 * 256];
#pragma unroll
    for (int v = 0; v < 8; ++v) red[wave * 256 + v * 32 + lane] = c0[v];
    __syncthreads();
    {
        const int v = tid >> 5;
        const int l = tid & 31;
        float s = 0.0f;
#pragma unroll
        for (int w = 0; w < 8; ++w) s += red[w * 256 + v * 32 + l];
        const int rl   = v + (l >> 4) * 8;
        const int b    = l & 15;
        const int orow = tile * 16 + rl;
        out[orow * BATCH + b] = scales[orow] * s + bias[orow];
    }
}

extern "C" void kernel_launch(void* const* d_in, const int* in_sizes, int n_in,
                              void* d_out, int out_size, void* d_ws, size_t ws_size,
                              hipStream_t stream) {
    const float* x    = (const float*)d_in[0];   // (16, 8192)
    const float* W    = (const float*)d_in[1];   // (8192, 8192)
    const float* bias = (const float*)d_in[2];   // (8192,)
    float* out = (float*)d_out;                  // (8192, 16)

    float*  scales = (float*)d_ws;                       // 32 KB
    int8_t* wq     = (int8_t*)d_ws + 32768;              // 64 MB
    const size_t need = 32768 + (size_t)OUT_F * IN_F;
    const int use_q = (ws_size >= need) ? 1 : 0;         // deterministic in ws_size

    quant_rows_kernel<<<OUT_F, 256, 0, stream>>>(W, scales, wq, use_q);
    if (use_q) {
        gemm_wmma_q_kernel<<<OUT_F / 32, 256, 0, stream>>>(wq, scales, x, bias, out);
    } else {
        gemm_wmma_f_kernel<<<OUT_F / 16, 256, 0, stream>>>(W, scales, x, bias, out);
    }
}